// DPQEmbedding_33346126086311
// MI455X (gfx1250) — compile-verified
//
#include <hip/hip_runtime.h>

typedef float v2f __attribute__((ext_vector_type(2)));
typedef float v8f __attribute__((ext_vector_type(8)));

#define EMB   128
#define D     8
#define K     128
#define SUB   16
#define NTOK  (1024 * 128)
#define NTILES (NTOK / 16)      // 8192 token tiles of 16
#define NSAMP  ((double)NTOK * (double)D)
#define BN_EPS 1e-3f

__device__ __forceinline__ v2f mk2(float a, float b) { v2f r; r[0] = a; r[1] = b; return r; }

// ---------------------------------------------------------------------------
// k0: zero f64 stat accumulators, precompute norm_c[d][k] = sum_s c^2
// ---------------------------------------------------------------------------
__global__ __launch_bounds__(1024) void k0_init(const float* __restrict__ cent,
                                                double* __restrict__ sums,    // 256 doubles
                                                float* __restrict__ norm_c) { // 1024 floats
    int i = threadIdx.x;
    if (i < 256) sums[i] = 0.0;
    const float* c = cent + i * SUB;
    float acc = 0.f;
#pragma unroll
    for (int s = 0; s < SUB; ++s) acc += c[s] * c[s];
    norm_c[i] = acc;
}

// ---------------------------------------------------------------------------
// k1: per-k sum/sumsq of response. Orientation: A = x (M=token), B = 2*cent.
// resp = seedWMMA(-nx[m]-nc[n]) + x·(2c)  — no broadcasts, no dynamic indexing.
// ---------------------------------------------------------------------------
__global__ __launch_bounds__(256) void k1_stats(const int* __restrict__ ids,
                                                const float* __restrict__ wemb,
                                                const float* __restrict__ cent,
                                                const float* __restrict__ norm_c,
                                                double* __restrict__ gsum,
                                                double* __restrict__ gsumsq) {
    __shared__ float s_sum[K];
    __shared__ float s_sq[K];
    const int tid = threadIdx.x;
    if (tid < K) { s_sum[tid] = 0.f; s_sq[tid] = 0.f; }
    __syncthreads();

    const int lane = tid & 31;
    const int d    = tid >> 5;        // wave id == subspace
    const int half = lane >> 4;
    const int col  = lane & 15;       // B-side: k within tile; A-side: token within tile

    const float* cd = cent + (size_t)d * K * SUB;
    v2f Bc[8][4];                     // centroid fragments, pre-scaled by 2
    v2f Bs[8];                        // seed-B: rows (-1, -nc[k])
    float lsum[8], lsq[8];
#pragma unroll
    for (int kt = 0; kt < 8; ++kt) {
        const int k = kt * 16 + col;
        const float* ck = cd + (size_t)k * SUB;
#pragma unroll
        for (int kk = 0; kk < 4; ++kk) {
            const int s = kk * 4 + half * 2;
            Bc[kt][kk] = mk2(2.f * ck[s], 2.f * ck[s + 1]);
        }
        Bs[kt]   = half ? mk2(0.f, 0.f) : mk2(-1.f, -norm_c[d * K + k]);
        lsum[kt] = 0.f;
        lsq[kt]  = 0.f;
    }

    for (int tile = blockIdx.x; tile < NTILES; tile += gridDim.x) {
        const int tok = tile * 16 + col;
        const float* xr = wemb + (size_t)ids[tok] * EMB + d * SUB;
        v2f Xf[4];
        float p = 0.f;
#pragma unroll
        for (int kk = 0; kk < 4; ++kk) {
            const int s = kk * 4 + half * 2;        // lane-halves split the 16-float row
            Xf[kk] = *(const v2f*)(xr + s);
            p = fmaf(Xf[kk][0], Xf[kk][0], p);
            p = fmaf(Xf[kk][1], Xf[kk][1], p);
        }
        const float nx = p + __shfl_xor(p, 16, 32); // complete the split ||x||^2
        const v2f A2 = half ? mk2(0.f, 0.f) : mk2(nx, 1.f);

#pragma unroll
        for (int kt = 0; kt < 8; ++kt) {
            v8f z = {0.f,0.f,0.f,0.f,0.f,0.f,0.f,0.f};
            // seed: C[m,n] = -nx[m] - nc[n]
            v8f acc = __builtin_amdgcn_wmma_f32_16x16x4_f32(
                false, A2, false, Bs[kt], (short)0, z, false, false);
#pragma unroll
            for (int kk = 0; kk < 4; ++kk)
                acc = __builtin_amdgcn_wmma_f32_16x16x4_f32(
                    false, Xf[kk], false, Bc[kt][kk], (short)0, acc, false, false);
#pragma unroll
            for (int r = 0; r < 8; ++r) {
                lsum[kt] += acc[r];
                lsq[kt]   = fmaf(acc[r], acc[r], lsq[kt]);
            }
        }
    }

#pragma unroll
    for (int kt = 0; kt < 8; ++kt) {
        atomicAdd(&s_sum[kt * 16 + col], lsum[kt]);
        atomicAdd(&s_sq [kt * 16 + col], lsq [kt]);
    }
    __syncthreads();
    if (tid < K) {
        atomicAdd(&gsum  [tid], (double)s_sum[tid]);
        atomicAdd(&gsumsq[tid], (double)s_sq [tid]);
    }
}

// ---------------------------------------------------------------------------
// k2: mean[k], inv_std[k]
// ---------------------------------------------------------------------------
__global__ __launch_bounds__(128) void k2_finalize(const double* __restrict__ gsum,
                                                   const double* __restrict__ gsumsq,
                                                   float* __restrict__ mean,
                                                   float* __restrict__ inv_std) {
    int k = threadIdx.x;
    double m = gsum[k] / NSAMP;
    double v = gsumsq[k] / NSAMP - m * m;
    mean[k]    = (float)m;
    inv_std[k] = rsqrtf((float)v + BN_EPS);
}

// ---------------------------------------------------------------------------
// k3: argmax pass. Swapped orientation: A = (2*is[k])*cent (M=k), B = x^T
// (N=token). bn value arrives directly in acc via affine-seeded WMMA:
//   acc[m,n] = seed(-beta[m] - is[m]*nx[n]) + (2*is[m]*c[m])·x[n]
// One token per lane -> per-lane exact argmax over 64 k's, then 1 shfl_xor(16).
// ---------------------------------------------------------------------------
__global__ __launch_bounds__(256) void k3_argmax(const int* __restrict__ ids,
                                                 const float* __restrict__ wemb,
                                                 const float* __restrict__ cent,
                                                 const float* __restrict__ norm_c,
                                                 const float* __restrict__ mean,
                                                 const float* __restrict__ inv_std,
                                                 float* __restrict__ out) {
    const int tid  = threadIdx.x;
    const int lane = tid & 31;
    const int d    = tid >> 5;
    const int half = lane >> 4;
    const int col  = lane & 15;       // A-side: k within tile; B-side: token within tile

    const float* cd = cent + (size_t)d * K * SUB;
    v2f Ac[8][4];                     // centroid fragments pre-scaled by alpha = 2*is[k]
    v2f A2[8];                        // seed-A: (beta[k], is[k])
#pragma unroll
    for (int kt = 0; kt < 8; ++kt) {
        const int k = kt * 16 + col;
        const float isv   = inv_std[k];
        const float alpha = 2.f * isv;
        const float beta  = isv * (norm_c[d * K + k] + mean[k]);
        const float* ck = cd + (size_t)k * SUB;
#pragma unroll
        for (int kk = 0; kk < 4; ++kk) {
            const int s = kk * 4 + half * 2;
            Ac[kt][kk] = mk2(alpha * ck[s], alpha * ck[s + 1]);
        }
        A2[kt] = half ? mk2(0.f, 0.f) : mk2(beta, isv);
    }

    for (int tile = blockIdx.x; tile < NTILES; tile += gridDim.x) {
        const int tok = tile * 16 + col;
        const float* xr = wemb + (size_t)ids[tok] * EMB + d * SUB;
        v2f Bx[4];
        float p = 0.f;
#pragma unroll
        for (int kk = 0; kk < 4; ++kk) {
            const int s = kk * 4 + half * 2;
            Bx[kk] = *(const v2f*)(xr + s);
            p = fmaf(Bx[kk][0], Bx[kk][0], p);
            p = fmaf(Bx[kk][1], Bx[kk][1], p);
        }
        const float nx = p + __shfl_xor(p, 16, 32);
        const v2f B2 = half ? mk2(0.f, 0.f) : mk2(-1.f, -nx);

        float bestv = -3.4e38f;
        int   bestk = 0;
#pragma unroll
        for (int kt = 0; kt < 8; ++kt) {
            v8f z = {0.f,0.f,0.f,0.f,0.f,0.f,0.f,0.f};
            v8f acc = __builtin_amdgcn_wmma_f32_16x16x4_f32(
                false, A2[kt], false, B2, (short)0, z, false, false);
#pragma unroll
            for (int kk = 0; kk < 4; ++kk)
                acc = __builtin_amdgcn_wmma_f32_16x16x4_f32(
                    false, Ac[kt][kk], false, Bx[kk], (short)0, acc, false, false);
#pragma unroll
            for (int r = 0; r < 8; ++r) {
                const int kidx = kt * 16 + 8 * half + r;   // ascending per lane
                if (acc[r] > bestv) { bestv = acc[r]; bestk = kidx; }
            }
        }

        // merge the two lane-halves (k mod 16 in 0..7 vs 8..15), min-k tie-break
        {
            const float ov = __shfl_xor(bestv, 16, 32);
            const int   ok = __shfl_xor(bestk, 16, 32);
            if (ov > bestv || (ov == bestv && ok < bestk)) { bestv = ov; bestk = ok; }
        }

        // lane owns token `tok`: emit 8 of its 16 floats (halves split the row)
        const float* src = cd + (size_t)bestk * SUB + half * 8;
        float4 c0 = *(const float4*)(src);
        float4 c1 = *(const float4*)(src + 4);
        float* dst = out + (size_t)tok * EMB + d * SUB + half * 8;
        *(float4*)(dst)     = c0;
        *(float4*)(dst + 4) = c1;
    }
}

// ---------------------------------------------------------------------------
extern "C" void kernel_launch(void* const* d_in, const int* in_sizes, int n_in,
                              void* d_out, int out_size, void* d_ws, size_t ws_size,
                              hipStream_t stream) {
    const int*   ids  = (const int*)d_in[0];     // (1024,128) int32
    const float* wemb = (const float*)d_in[1];   // (100000,128) f32
    const float* cent = (const float*)d_in[2];   // (8,128,16) f32
    float* out = (float*)d_out;                  // (1024,128,128) f32

    double* gsum   = (double*)d_ws;      // [128]
    double* gsumsq = gsum + 128;         // [128]
    float*  mean   = (float*)(gsum + 256);
    float*  invstd = mean + 128;
    float*  normc  = invstd + 128;       // [8][128]

    k0_init    <<<1,    1024, 0, stream>>>(cent, gsum, normc);
    k1_stats   <<<1024, 256,  0, stream>>>(ids, wemb, cent, normc, gsum, gsumsq);
    k2_finalize<<<1,    128,  0, stream>>>(gsum, gsumsq, mean, invstd);
    k3_argmax  <<<2048, 256,  0, stream>>>(ids, wemb, cent, normc, mean, invstd, out);
}